// HAN_7335804141697
// MI455X (gfx1250) — compile-verified
//
#include <hip/hip_runtime.h>
#include <hip/hip_bf16.h>
#include <math.h>

// ---------------------------------------------------------------------------
// HAN forward for MI455X (gfx1250, wave32).
// Dense GEMMs -> V_WMMA_F32_16X16X4_F32, one 16-row strip per wave with
// A-operand reuse across all column tiles (1 A-load feeds 4..8 WMMAs).
// Edge softmax/aggregation -> float CAS-max + global atomicAdd, attn vectors
// staged in LDS.
// ---------------------------------------------------------------------------

#define N_NODES 100000
#define T_NODES 40000
#define E_EDGES 500000
#define D_DIM   64
#define H_HEADS 8
#define SEM_DIM 128
#define NUM_CLS 8

typedef __attribute__((ext_vector_type(2))) float v2f;
typedef __attribute__((ext_vector_type(8))) float v8f;

__device__ __forceinline__ v8f wmma_f32_4(v2f a, v2f b, v8f c) {
  // 8 args: (neg_a, A, neg_b, B, c_mod, C, reuse_a, reuse_b)
  return __builtin_amdgcn_wmma_f32_16x16x4_f32(
      false, a, false, b, (short)0, c, false, false);
}

__device__ __forceinline__ void atomicMaxF(float* addr, float val) {
  unsigned int* u = (unsigned int*)addr;
  unsigned int old = __float_as_uint(*addr);
  while (__uint_as_float(old) < val) {
    unsigned int assumed = old;
    old = atomicCAS(u, assumed, __float_as_uint(val));
    if (old == assumed) break;
  }
}

// --------------------------- utility fills ---------------------------------
__global__ void fill_kernel(float* __restrict__ p, float v, int n) {
  int i = blockIdx.x * blockDim.x + threadIdx.x;
  if (i < n) p[i] = v;
}

// ------------------- feature projection: feat @ W^T + b --------------------
// One wave computes a full 16x64 output strip (4 column tiles), so each
// A-element load is reused by 4 WMMAs.
__global__ void proj_wmma_kernel(const float* __restrict__ feat,
                                 const float* __restrict__ W,
                                 const float* __restrict__ bias,
                                 float* __restrict__ out,
                                 int C, int R, int rowOffset) {
  int wave = (blockIdx.x * blockDim.x + threadIdx.x) >> 5;
  int lane = threadIdx.x & 31;
  int strips = C / 16;
  if (wave >= strips) return;  // wave-uniform
  int row0 = wave * 16;
  int half = lane >> 4;   // K sub-group (K pairs 0-1 vs 2-3)
  int idx16 = lane & 15;  // M (for A) / N (for B)

  const float* arow = feat + (size_t)(row0 + idx16) * R + 2 * half;
  const float* brow0 = W + (size_t)(0 * 16 + idx16) * R + 2 * half;
  const float* brow1 = W + (size_t)(1 * 16 + idx16) * R + 2 * half;
  const float* brow2 = W + (size_t)(2 * 16 + idx16) * R + 2 * half;
  const float* brow3 = W + (size_t)(3 * 16 + idx16) * R + 2 * half;

  v8f c0 = {}, c1 = {}, c2 = {}, c3 = {};
  for (int k = 0; k < R; k += 4) {
    if ((k & 127) == 0) __builtin_prefetch(arow + k + 256, 0, 0);
    v2f a, b0, b1, b2, b3;
    a.x = arow[k];      a.y = arow[k + 1];
    b0.x = brow0[k];    b0.y = brow0[k + 1];
    b1.x = brow1[k];    b1.y = brow1[k + 1];
    b2.x = brow2[k];    b2.y = brow2[k + 1];
    b3.x = brow3[k];    b3.y = brow3[k + 1];
    c0 = wmma_f32_4(a, b0, c0);
    c1 = wmma_f32_4(a, b1, c1);
    c2 = wmma_f32_4(a, b2, c2);
    c3 = wmma_f32_4(a, b3, c3);
  }
  int outRow0 = rowOffset + row0;
#pragma unroll
  for (int v = 0; v < 8; ++v) {
    float* orow = out + (size_t)(outRow0 + v + 8 * half) * D_DIM;
    orow[0 * 16 + idx16] = c0[v] + bias[0 * 16 + idx16];
    orow[1 * 16 + idx16] = c1[v] + bias[1 * 16 + idx16];
    orow[2 * 16 + idx16] = c2[v] + bias[2 * 16 + idx16];
    orow[3 * 16 + idx16] = c3[v] + bias[3 * 16 + idx16];
  }
}

// ---------------- edge scores: leaky-relu(concat(h_dst,h_src)·a_h) ---------
// wave per edge; lanes 0-15 cover h_dst dims, lanes 16-31 cover h_src dims.
// Attention table staged in LDS once per block.
__global__ void edge_score_kernel(const float* __restrict__ nf,
                                  const int* __restrict__ mp,
                                  const float* __restrict__ attn,  // [8][128]
                                  float* __restrict__ score,       // [E,8]
                                  float* __restrict__ mmax) {      // [N,8]
  __shared__ float sattn[H_HEADS * 2 * D_DIM];  // 4 KB
  for (int i = threadIdx.x; i < H_HEADS * 2 * D_DIM; i += blockDim.x)
    sattn[i] = attn[i];
  __syncthreads();

  int wave = (blockIdx.x * blockDim.x + threadIdx.x) >> 5;
  int lane = threadIdx.x & 31;
  if (wave >= E_EDGES) return;
  int src = mp[wave * 2 + 0];
  int dst = mp[wave * 2 + 1];
  int node = (lane < 16) ? dst : src;
  const float4 ev =
      *(const float4*)(nf + (size_t)node * D_DIM + ((lane * 4) & 63));
#pragma unroll
  for (int h = 0; h < H_HEADS; ++h) {
    float4 av = *(const float4*)(sattn + h * 2 * D_DIM + lane * 4);
    float p = ev.x * av.x + ev.y * av.y + ev.z * av.z + ev.w * av.w;
#pragma unroll
    for (int off = 16; off >= 1; off >>= 1) p += __shfl_xor(p, off, 32);
    if (lane == 0) {
      p = (p > 0.f) ? p : 0.01f * p;  // leaky relu
      score[(size_t)wave * H_HEADS + h] = p;
      atomicMaxF(mmax + (size_t)dst * H_HEADS + h, p);
    }
  }
}

// --------------- exp(score - max[dst]) and segment denominator -------------
__global__ void edge_exp_kernel(float* __restrict__ score,  // in: score, out: ex
                                const int* __restrict__ mp,
                                const float* __restrict__ mmax,
                                float* __restrict__ den) {
  int idx = blockIdx.x * blockDim.x + threadIdx.x;
  if (idx >= E_EDGES * H_HEADS) return;
  int e = idx >> 3, h = idx & 7;
  int dst = mp[e * 2 + 1];
  float ex = expf(score[idx] - mmax[dst * H_HEADS + h]);
  score[idx] = ex;
  atomicAdd(den + dst * H_HEADS + h, ex);
}

// ------------- weighted scatter-add per head: agg[dst] += attn*h_src -------
__global__ void edge_agg_kernel(const float* __restrict__ nf,
                                const int* __restrict__ mp,
                                const float* __restrict__ ex,
                                const float* __restrict__ den,
                                float* __restrict__ aggh,  // [N,64]
                                int head) {
  int wave = (blockIdx.x * blockDim.x + threadIdx.x) >> 5;
  int lane = threadIdx.x & 31;
  if (wave >= E_EDGES) return;
  int src = mp[wave * 2 + 0];
  int dst = mp[wave * 2 + 1];
  float w = ex[(size_t)wave * H_HEADS + head] / den[dst * H_HEADS + head];
  const float* hs = nf + (size_t)src * D_DIM;
  float* ag = aggh + (size_t)dst * D_DIM;
  atomicAdd(ag + lane, hs[lane] * w);
  atomicAdd(ag + lane + 32, hs[lane + 32] * w);
}

// --------------------- gather aggregates at target nodes -------------------
__global__ void gather_kernel(const float* __restrict__ aggh,
                              const int* __restrict__ tgt,
                              float* __restrict__ hout,  // [T, 512]
                              int head) {
  int idx = blockIdx.x * blockDim.x + threadIdx.x;
  if (idx >= T_NODES * D_DIM) return;
  int t = idx >> 6, d = idx & 63;
  hout[(size_t)t * (H_HEADS * D_DIM) + head * D_DIM + d] =
      aggh[(size_t)tgt[t] * D_DIM + d];
}

// ----- semantic scores: sum_t sum_s tanh(h@Wsem^T + bsem)*a_sem  (WMMA) ----
// One wave computes a full 16x128 strip (8 column tiles) so each A-load
// feeds 8 WMMAs; result reduced directly to the two semantic scalars.
__global__ void sem_score_kernel(const float* __restrict__ h,     // [2,T,512]
                                 const float* __restrict__ Wsem,  // [128,512]
                                 const float* __restrict__ bsem,
                                 const float* __restrict__ a_sem,
                                 float* __restrict__ sred) {
  int wave = (blockIdx.x * blockDim.x + threadIdx.x) >> 5;
  int lane = threadIdx.x & 31;
  const int K = H_HEADS * D_DIM;  // 512
  const int strips = T_NODES / 16;  // 2500 per metapath
  int total = 2 * strips;
  if (wave >= total) return;  // wave-uniform
  int m = wave / strips;
  int tt = wave % strips;
  int half = lane >> 4, idx16 = lane & 15;

  const float* arow =
      h + (size_t)m * T_NODES * K + (size_t)(tt * 16 + idx16) * K + 2 * half;
  const float* brow[8];
#pragma unroll
  for (int j = 0; j < 8; ++j)
    brow[j] = Wsem + (size_t)(j * 16 + idx16) * K + 2 * half;

  v8f acc[8];
#pragma unroll
  for (int j = 0; j < 8; ++j) acc[j] = (v8f){};

  for (int k = 0; k < K; k += 4) {
    if ((k & 127) == 0) __builtin_prefetch(arow + k + 256, 0, 0);
    v2f a;
    a.x = arow[k];
    a.y = arow[k + 1];
#pragma unroll
    for (int j = 0; j < 8; ++j) {
      v2f b;
      b.x = brow[j][k];
      b.y = brow[j][k + 1];
      acc[j] = wmma_f32_4(a, b, acc[j]);
    }
  }

  float s = 0.f;
#pragma unroll
  for (int j = 0; j < 8; ++j) {
    int col = j * 16 + idx16;
    float bv = bsem[col], av = a_sem[col];
#pragma unroll
    for (int v = 0; v < 8; ++v) s += tanhf(acc[j][v] + bv) * av;
  }
#pragma unroll
  for (int off = 16; off >= 1; off >>= 1) s += __shfl_xor(s, off, 32);
  if (lane == 0) atomicAdd(sred + m, s);
}

// ----------------------- beta = softmax(scores / T) ------------------------
__global__ void beta_kernel(float* __restrict__ sred, float invT) {
  float s0 = sred[0] * invT, s1 = sred[1] * invT;
  float mx = fmaxf(s0, s1);
  float e0 = expf(s0 - mx), e1 = expf(s1 - mx);
  float inv = 1.f / (e0 + e1);
  sred[2] = e0 * inv;
  sred[3] = e1 * inv;
}

// ------------------- embeddings = beta0*h0 + beta1*h1 ----------------------
__global__ void combine_kernel(const float* __restrict__ h,
                               const float* __restrict__ sred,
                               float* __restrict__ emb, int n) {
  int i = blockIdx.x * blockDim.x + threadIdx.x;
  if (i >= n) return;
  emb[i] = sred[2] * h[i] + sred[3] * h[(size_t)n + i];
}

// --------------------- cls_logits = emb @ Wcls^T + bcls --------------------
__global__ void cls_kernel(const float* __restrict__ emb,
                           const float* __restrict__ Wc,
                           const float* __restrict__ bc,
                           float* __restrict__ logits) {
  int idx = blockIdx.x * blockDim.x + threadIdx.x;
  if (idx >= T_NODES * NUM_CLS) return;
  int t = idx >> 3, c = idx & 7;
  const float* e = emb + (size_t)t * (H_HEADS * D_DIM);
  const float* w = Wc + (size_t)c * (H_HEADS * D_DIM);
  float s = bc[c];
  for (int k = 0; k < H_HEADS * D_DIM; k += 4) {
    s += e[k] * w[k] + e[k + 1] * w[k + 1] + e[k + 2] * w[k + 2] +
         e[k + 3] * w[k + 3];
  }
  logits[idx] = s;
}

// ---------------------------------------------------------------------------
extern "C" void kernel_launch(void* const* d_in, const int* in_sizes, int n_in,
                              void* d_out, int out_size, void* d_ws,
                              size_t ws_size, hipStream_t stream) {
  const int* target = (const int*)d_in[0];
  const int* mp0 = (const int*)d_in[1];
  const int* mp1 = (const int*)d_in[2];
  // d_in[3] node_type_mapping: implied by concatenation order, unused.
  const float* feat0 = (const float*)d_in[4];
  const float* W0 = (const float*)d_in[5];
  const float* b0 = (const float*)d_in[6];
  const float* feat1 = (const float*)d_in[7];
  const float* W1 = (const float*)d_in[8];
  const float* b1 = (const float*)d_in[9];
  const float* feat2 = (const float*)d_in[10];
  const float* W2 = (const float*)d_in[11];
  const float* b2 = (const float*)d_in[12];
  const float* nattn = (const float*)d_in[13];  // [2,8,128]
  const float* Wsem = (const float*)d_in[14];
  const float* bsem = (const float*)d_in[15];
  const float* a_sem = (const float*)d_in[16];
  const float* Wcls = (const float*)d_in[17];
  const float* bcls = (const float*)d_in[18];

  float* ws = (float*)d_ws;
  size_t off = 0;
  float* nf = ws + off;   off += (size_t)N_NODES * D_DIM;               // 6.4M
  float* exb = ws + off;  off += (size_t)E_EDGES * H_HEADS;             // 4.0M
  float* mmax = ws + off; off += (size_t)N_NODES * H_HEADS;             // 0.8M
  float* den = ws + off;  off += (size_t)N_NODES * H_HEADS;             // 0.8M
  float* aggh = ws + off; off += (size_t)N_NODES * D_DIM;               // 6.4M
  float* hbuf = ws + off; off += (size_t)2 * T_NODES * H_HEADS * D_DIM; // 41M
  float* sred = ws + off; off += 4;

  float* out_logits = (float*)d_out;
  float* out_emb = out_logits + (size_t)T_NODES * NUM_CLS;

  const int B = 256;
  auto wblocks = [](int waves) { return (waves + 7) / 8; };  // 8 waves/block

  // 1) feature projections (WMMA f32 16x16x4, 16x64 strip per wave)
  proj_wmma_kernel<<<wblocks(40000 / 16), B, 0, stream>>>(feat0, W0, b0, nf,
                                                          40000, 512, 0);
  proj_wmma_kernel<<<wblocks(40000 / 16), B, 0, stream>>>(feat1, W1, b1, nf,
                                                          40000, 256, 40000);
  proj_wmma_kernel<<<wblocks(20000 / 16), B, 0, stream>>>(feat2, W2, b2, nf,
                                                          20000, 128, 80000);

  // 2) metapath attention + aggregation
  for (int m = 0; m < 2; ++m) {
    const int* mp = m ? mp1 : mp0;
    const float* attn = nattn + (size_t)m * H_HEADS * 2 * D_DIM;
    fill_kernel<<<(N_NODES * H_HEADS + B - 1) / B, B, 0, stream>>>(
        mmax, -INFINITY, N_NODES * H_HEADS);
    fill_kernel<<<(N_NODES * H_HEADS + B - 1) / B, B, 0, stream>>>(
        den, 0.f, N_NODES * H_HEADS);
    edge_score_kernel<<<wblocks(E_EDGES), B, 0, stream>>>(nf, mp, attn, exb,
                                                          mmax);
    edge_exp_kernel<<<(E_EDGES * H_HEADS + B - 1) / B, B, 0, stream>>>(
        exb, mp, mmax, den);
    for (int h = 0; h < H_HEADS; ++h) {
      fill_kernel<<<(N_NODES * D_DIM + B - 1) / B, B, 0, stream>>>(
          aggh, 0.f, N_NODES * D_DIM);
      edge_agg_kernel<<<wblocks(E_EDGES), B, 0, stream>>>(nf, mp, exb, den,
                                                          aggh, h);
      gather_kernel<<<(T_NODES * D_DIM + B - 1) / B, B, 0, stream>>>(
          aggh, target, hbuf + (size_t)m * T_NODES * H_HEADS * D_DIM, h);
    }
  }

  // 3) semantic attention (WMMA) -> beta -> combine -> classifier
  fill_kernel<<<1, B, 0, stream>>>(sred, 0.f, 4);
  sem_score_kernel<<<wblocks(2 * (T_NODES / 16)), B, 0, stream>>>(
      hbuf, Wsem, bsem, a_sem, sred);
  beta_kernel<<<1, 1, 0, stream>>>(sred, 1.0f / (float)T_NODES);
  combine_kernel<<<(T_NODES * H_HEADS * D_DIM + B - 1) / B, B, 0, stream>>>(
      hbuf, sred, out_emb, T_NODES * H_HEADS * D_DIM);
  cls_kernel<<<(T_NODES * NUM_CLS + B - 1) / B, B, 0, stream>>>(
      out_emb, Wcls, bcls, out_logits);
}